// MyModel_61933428415156
// MI455X (gfx1250) — compile-verified
//
#include <hip/hip_runtime.h>

typedef float v2f __attribute__((ext_vector_type(2)));
typedef float v8f __attribute__((ext_vector_type(8)));

#define LOG2E 1.44269504088896340736f

// half-wave swap (lane L <-> lane L^16) without touching the LDS pipeline if
// v_permlanex16 is available; otherwise one ds_swizzle (group-of-32 xor 16).
__device__ __forceinline__ float xor16(float x) {
  int i = __float_as_int(x);
#if __has_builtin(__builtin_amdgcn_permlanex16)
  i = __builtin_amdgcn_permlanex16(i, i, 0x76543210, 0xfedcba98, true, false);
#else
  i = __builtin_amdgcn_ds_swizzle(i, 0x401f);  // xor=0x10, and=0x1f
#endif
  return __int_as_float(i);
}

__device__ __forceinline__ v8f wmma4(v2f a, v2f b, v8f c) {
  // D = A(16x4 f32) * B(4x16 f32) + C(16x16 f32)
  return __builtin_amdgcn_wmma_f32_16x16x4_f32(false, a, false, b, (short)0, c,
                                               false, false);
}

// sigma(x) from a pre-scaled argument z = -log2e * x (scaling folded into
// the weights): sigma = rcp(1 + 2^z)
__device__ __forceinline__ float sig_pre(float z) {
  return __builtin_amdgcn_rcpf(1.0f + __builtin_amdgcn_exp2f(z));
}

// One wave (32 lanes) owns 16 batch rows and scans the whole sequence for one
// direction. Per step: z^T = [Wih|Whh] @ [x_t|h]^T + b via 8 chained
// v_wmma_f32_16x16x4_f32 (weights live in VGPRs as the A operand all scan),
// then gate nonlinearities + state update. Both wave halves mirror the same
// 16 batch rows, so after ONE half-swap of the activated gates each half can
// compute c/h locally and the next step's B operand is pure lane-local selects.
__global__ __launch_bounds__(32) void lstm_scan_kernel(
    const float* __restrict__ xin,
    const float* __restrict__ WihF, const float* __restrict__ WhhF,
    const float* __restrict__ bF,
    const float* __restrict__ WihB, const float* __restrict__ WhhB,
    const float* __restrict__ bB,
    float* __restrict__ outF, float* __restrict__ outB, int T) {
  const int lane = threadIdx.x;      // 0..31
  const int dir  = blockIdx.x & 1;   // 0 = fwd, 1 = bwd
  const int grp  = blockIdx.x >> 1;  // 16 batch rows per group
  const int lm   = lane & 15;
  const bool low = lane < 16;
  const int row  = grp * 16 + lm;

  const float* Wih = dir ? WihB : WihF;
  const float* Whh = dir ? WhhB : WhhF;
  const float* bb  = dir ? bB : bF;
  float* out       = dir ? outB : outF;

  // --- A operands (weights) in WMMA A-layout, resident in VGPRs all scan ---
  // A tile t covers gates 16t..16t+15; K chunks 0,1 = Wih cols, 2,3 = Whh.
  // Rows are pre-scaled by -log2e (i,f,o gates) or -2*log2e (g gates) so the
  // activations become rcp(1 + exp2(z)) with no per-step input scaling.
  v2f A[2][4];
  v8f bias[2];
  const int koff = low ? 0 : 2;
#pragma unroll
  for (int t0 = 0; t0 < 2; ++t0) {
    const int g = 16 * t0 + lm;
    const float kA = (g >= 16 && g < 24) ? (-2.f * LOG2E) : (-LOG2E);
#pragma unroll
    for (int c = 0; c < 2; ++c) {
      A[t0][c].x     = Wih[g * 8 + 4 * c + koff] * kA;
      A[t0][c].y     = Wih[g * 8 + 4 * c + koff + 1] * kA;
      A[t0][2 + c].x = Whh[g * 8 + 4 * c + koff] * kA;
      A[t0][2 + c].y = Whh[g * 8 + 4 * c + koff + 1] * kA;
    }
    // Bias in C layout: VGPR v <-> gate 16t + v (+8 on the upper half-wave).
    const int gb = 16 * t0 + (low ? 0 : 8);
#pragma unroll
    for (int v = 0; v < 8; ++v) {
      const int gg = gb + v;
      const float kB = (gg >= 16 && gg < 24) ? (-2.f * LOG2E) : (-LOG2E);
      bias[t0][v] = bb[gg] * kB;
    }
  }

  // Tile1 mixed activation: lanes 0-15 carry g-gates (tanh = 2*sig-1),
  // lanes 16-31 carry o-gates (plain sigmoid).
  const float m1 = low ? 2.f : 1.f;
  const float a1 = low ? -1.f : 0.f;

  const float4* xr = (const float4*)(xin + (size_t)row * T * 8);
  float4* orow     = (float4*)(out + (size_t)row * T * 8);

  float c[8], h[8];
#pragma unroll
  for (int v = 0; v < 8; ++v) { c[v] = 0.f; h[v] = 0.f; }

  const int tstep = dir ? -1 : 1;
  int t = dir ? (T - 1) : 0;
  // software-pipelined x fetch (double buffer; next index clamped branchless)
  float4 xlo = xr[2 * t];
  float4 xhi = xr[2 * t + 1];

  for (int s = 0; s < T; ++s) {
    // B operands (x part): upper half mirrors the same rows, so both halves
    // already hold the right values -> pure selects.
    v2f bx0, bx1;
    bx0.x = low ? xlo.x : xlo.z;  bx0.y = low ? xlo.y : xlo.w;
    bx1.x = low ? xhi.x : xhi.z;  bx1.y = low ? xhi.y : xhi.w;

    // prefetch next timestep's x (clamped; redundant last load is harmless)
    int tn = t + tstep;
    tn = (tn < 0) ? 0 : ((tn >= T) ? (T - 1) : tn);
    const float4 nlo = xr[2 * tn];
    const float4 nhi = xr[2 * tn + 1];

    // B operands (h part): h is valid on ALL lanes -> lane-local selects.
    v2f bh0, bh1;
    bh0.x = low ? h[0] : h[2];  bh0.y = low ? h[1] : h[3];
    bh1.x = low ? h[4] : h[6];  bh1.y = low ? h[5] : h[7];

    // z^T tiles: tile0 = gates i(0-7),f(8-15); tile1 = g(16-23),o(24-31)
    v8f z0 = bias[0];
    z0 = wmma4(A[0][0], bx0, z0);
    z0 = wmma4(A[0][1], bx1, z0);
    z0 = wmma4(A[0][2], bh0, z0);
    z0 = wmma4(A[0][3], bh1, z0);
    v8f z1 = bias[1];
    z1 = wmma4(A[1][0], bx0, z1);
    z1 = wmma4(A[1][1], bx1, z1);
    z1 = wmma4(A[1][2], bh0, z1);
    z1 = wmma4(A[1][3], bh1, z1);

#pragma unroll
    for (int v = 0; v < 8; ++v) {
      // e0: lanes<16 sig(i) ; lanes>=16 sig(f)
      const float e0 = sig_pre(z0[v]);
      // act1: lanes<16 tanh(g) ; lanes>=16 sig(o)
      const float act1 = fmaf(m1, sig_pre(z1[v]), a1);
      // one half-swap gives each half the partner gates
      const float x0 = xor16(e0);    // lanes<16: sig(f) ; >=16: sig(i)
      const float x1 = xor16(act1);  // lanes<16: sig(o) ; >=16: tanh(g)
      const float si = low ? e0 : x0;
      const float sf = low ? x0 : e0;
      const float tg = low ? act1 : x1;
      const float so = low ? x1 : act1;
      c[v] = fmaf(sf, c[v], si * tg);               // c = f*c + i*tanh(g)
      const float tc =
          fmaf(2.f, sig_pre(c[v] * (-2.f * LOG2E)), -1.f);  // tanh(c)
      h[v] = so * tc;                               // h = o * tanh(c)
    }

    if (low) {  // both halves hold identical h; lower half stores
      orow[2 * t]     = make_float4(h[0], h[1], h[2], h[3]);
      orow[2 * t + 1] = make_float4(h[4], h[5], h[6], h[7]);
    }

    t = tn;
    xlo = nlo;
    xhi = nhi;
  }
}

// y[bt,:] = [ha[bt,:], hb[bt,:]] @ W^T + bias   (W: [8,16])
// Elementwise over 1M rows; weights staged in LDS (broadcast reads).
// Safe for out == ha (each thread reads its own row before writing it).
__global__ __launch_bounds__(256) void mix_kernel(
    const float* ha, const float* hb, const float* __restrict__ W,
    const float* __restrict__ bias, float* out, int n) {
  __shared__ float sW[128];
  __shared__ float sb[8];
  const int tid = threadIdx.x;
  if (tid < 128) sW[tid] = W[tid];
  if (tid < 8) sb[tid] = bias[tid];
  __syncthreads();
  const int idx = blockIdx.x * 256 + tid;
  if (idx >= n) return;
  const float4* pa = (const float4*)(ha + (size_t)idx * 8);
  const float4* pb = (const float4*)(hb + (size_t)idx * 8);
  const float4 a0 = pa[0], a1 = pa[1], b0 = pb[0], b1 = pb[1];
  const float hcat[16] = {a0.x, a0.y, a0.z, a0.w, a1.x, a1.y, a1.z, a1.w,
                          b0.x, b0.y, b0.z, b0.w, b1.x, b1.y, b1.z, b1.w};
  float o[8];
#pragma unroll
  for (int j = 0; j < 8; ++j) {
    float acc = sb[j];
#pragma unroll
    for (int k = 0; k < 16; ++k) acc = fmaf(hcat[k], sW[j * 16 + k], acc);
    o[j] = acc;
  }
  float4* po = (float4*)(out + (size_t)idx * 8);
  po[0] = make_float4(o[0], o[1], o[2], o[3]);
  po[1] = make_float4(o[4], o[5], o[6], o[7]);
}

extern "C" void kernel_launch(void* const* d_in, const int* in_sizes, int n_in,
                              void* d_out, int out_size, void* d_ws,
                              size_t ws_size, hipStream_t stream) {
  const float* x     = (const float*)d_in[0];
  const float* Wih1f = (const float*)d_in[1];
  const float* Whh1f = (const float*)d_in[2];
  const float* b1f   = (const float*)d_in[3];
  const float* Wih1b = (const float*)d_in[4];
  const float* Whh1b = (const float*)d_in[5];
  const float* b1b   = (const float*)d_in[6];
  const float* Wih2f = (const float*)d_in[7];
  const float* Whh2f = (const float*)d_in[8];
  const float* b2f   = (const float*)d_in[9];
  const float* Wih2b = (const float*)d_in[10];
  const float* Whh2b = (const float*)d_in[11];
  const float* b2b   = (const float*)d_in[12];
  const float* W1    = (const float*)d_in[13];
  const float* bl1   = (const float*)d_in[14];
  const float* W2    = (const float*)d_in[15];
  const float* bl2   = (const float*)d_in[16];

  const int B = 256;
  const int T = in_sizes[0] / (B * 8);
  const size_t seg = (size_t)B * T * 8;  // floats per [B,T,8] tensor
  float* r0 = (float*)d_ws;              // h1f -> y1
  float* r1 = r0 + seg;                  // h1b -> h2f
  float* r2 = r1 + seg;                  // h2b
  const int nBT = B * T;

  const dim3 scanGrid((B / 16) * 2), scanBlk(32);
  const dim3 mixGrid((nBT + 255) / 256), mixBlk(256);

  // layer 1 (fwd + bwd waves in one launch)
  lstm_scan_kernel<<<scanGrid, scanBlk, 0, stream>>>(
      x, Wih1f, Whh1f, b1f, Wih1b, Whh1b, b1b, r0, r1, T);
  // y1 = [h1f,h1b] @ W1^T + bl1   (in place over r0)
  mix_kernel<<<mixGrid, mixBlk, 0, stream>>>(r0, r1, W1, bl1, r0, nBT);
  // layer 2
  lstm_scan_kernel<<<scanGrid, scanBlk, 0, stream>>>(
      r0, Wih2f, Whh2f, b2f, Wih2b, Whh2b, b2b, r1, r2, T);
  // out = [h2f,h2b] @ W2^T + bl2
  mix_kernel<<<mixGrid, mixBlk, 0, stream>>>(r1, r2, W2, bl2, (float*)d_out,
                                             nBT);
}